// QWENWrapper_2207613190890
// MI455X (gfx1250) — compile-verified
//
#include <hip/hip_runtime.h>
#include <hip/hip_bf16.h>
#include <stdint.h>

#define NEW_T 16
#define NHEAD 16
#define KVH_  8
#define DHEAD 128
#define HDIM  1024
#define FF    3072
#define VOCAB 50257
#define SMAX_ 4096
#define PAST_ 2048
#define ENDLEN (PAST_ + NEW_T)   // 2064
#define NLAYER 8
#define KCH   8                  // K-split chunks for GEMMs
#define SCH   516                // seq-split chunk for attn output (4 * 516 = 2064)

typedef __attribute__((ext_vector_type(2))) float v2f;
typedef __attribute__((ext_vector_type(8))) float v8f;

static __device__ __forceinline__ v8f wmma4(v2f a, v2f b, v8f c) {
  // V_WMMA_F32_16X16X4_F32 : D = A(16x4) x B(4x16) + C(16x16)
  return __builtin_amdgcn_wmma_f32_16x16x4_f32(false, a, false, b, (short)0, c,
                                               false, false);
}

// ---------------- embed dequant ----------------
__global__ void embed_k(const int* __restrict__ ids, const uint8_t* __restrict__ e,
                        const float* __restrict__ sc, const float* __restrict__ zp,
                        float* __restrict__ h) {
  int idx = blockIdx.x * blockDim.x + threadIdx.x;   // 16*1024
  int m = idx >> 10, j = idx & 1023;
  int id = ids[m];
  h[idx] = (float)e[(size_t)id * HDIM + j] * sc[id] + zp[id];
}

// ---------------- row RMS norm ----------------
__global__ void rms_rows(const float* __restrict__ x, const float* __restrict__ w,
                         float* __restrict__ y, int ncols) {
  int row = blockIdx.x;
  const float* xr = x + (size_t)row * ncols;
  float ss = 0.f;
  for (int j = threadIdx.x; j < ncols; j += blockDim.x) { float v = xr[j]; ss += v * v; }
  __shared__ float red[256];
  red[threadIdx.x] = ss; __syncthreads();
  for (int s = 128; s > 0; s >>= 1) {
    if ((int)threadIdx.x < s) red[threadIdx.x] += red[threadIdx.x + s];
    __syncthreads();
  }
  float inv = rsqrtf(red[0] / (float)ncols + 1e-6f);
  for (int j = threadIdx.x; j < ncols; j += blockDim.x)
    y[(size_t)row * ncols + j] = w[j] * xr[j] * inv;
}

// ---- split-K M=16 WMMA GEMM partial: P[kc](16xN) = A(16, KCL) @ B(KCL, N) ----
// KCL is compile-time so the inner loop unrolls by 8 with clustered loads.
// grid: (N/16/8 blocks of 8 waves, KCH chunks in y)
template <int KCL>
__global__ void gemm16_splitk(const float* __restrict__ A, const float* __restrict__ B,
                              float* __restrict__ P, int N, int lda, int ldb) {
  const int lane = threadIdx.x & 31;
  const int ntile = blockIdx.x * (blockDim.x >> 5) + (threadIdx.x >> 5);
  const int kc = blockIdx.y;
  if (ntile * 16 >= N) return;                 // wave-uniform
  const int col  = ntile * 16 + (lane & 15);
  const int arow = lane & 15;
  const int koff = (lane >> 4) << 1;           // 0 or 2
  const int rbase = (lane >> 4) << 3;          // 0 or 8
  const int kbeg = kc * KCL;
  const float* Ar = A + (size_t)arow * lda + kbeg + koff;
  const float* Br = B + (size_t)(kbeg + koff) * ldb + col;
  v8f c = {};
#pragma unroll 8
  for (int kk = 0; kk < KCL; kk += 4) {
    v2f a, b;
    a.x = Ar[kk];
    a.y = Ar[kk + 1];
    b.x = Br[(size_t)kk * ldb];
    b.y = Br[(size_t)(kk + 1) * ldb];
    c = wmma4(a, b, c);
  }
  float* Pc = P + (size_t)kc * 16 * N;
#pragma unroll
  for (int r = 0; r < 8; ++r) Pc[(size_t)(r + rbase) * N + col] = c[r];
}

// ---- deterministic partial reduction: C = (acc?C:0) + sum_kc P[kc] ----
__global__ void reduce_partials(const float* __restrict__ P, float* __restrict__ C,
                                int total, int kchunks, int acc) {
  int idx = blockIdx.x * 256 + threadIdx.x;
  if (idx >= total) return;
  float s = acc ? C[idx] : 0.f;
  for (int kc = 0; kc < kchunks; ++kc) s += P[(size_t)kc * total + idx];
  C[idx] = s;
}

// ---------------- q-norm + RoPE (in place on q: 16 x NH*D) ----------------
__global__ void qnorm_rope(float* __restrict__ q, const float* __restrict__ qw,
                           const float* __restrict__ cosE, const float* __restrict__ sinE) {
  const int m = blockIdx.x >> 4, h = blockIdx.x & 15, d = threadIdx.x;
  __shared__ float red[128];
  __shared__ float sv[128];
  float v = q[(size_t)m * (NHEAD * DHEAD) + h * DHEAD + d];
  red[d] = v * v; __syncthreads();
  for (int s = 64; s > 0; s >>= 1) { if (d < s) red[d] += red[d + s]; __syncthreads(); }
  float inv = rsqrtf(red[0] / (float)DHEAD + 1e-6f);
  float qn = qw[d] * v * inv;
  sv[d] = qn; __syncthreads();
  float rot = (d < 64) ? -sv[d + 64] : sv[d - 64];
  const int pos = PAST_ + m;
  q[(size_t)m * (NHEAD * DHEAD) + h * DHEAD + d] =
      qn * cosE[(size_t)pos * DHEAD + d] + rot * sinE[(size_t)pos * DHEAD + d];
}

// ---- k-norm + RoPE -> knew (KVH,D,16); copy v -> vnew (KVH,16,D) ----
__global__ void kv_rope(const float* __restrict__ kb, const float* __restrict__ vb,
                        const float* __restrict__ kw, const float* __restrict__ cosE,
                        const float* __restrict__ sinE, float* __restrict__ knew,
                        float* __restrict__ vnew) {
  const int m = blockIdx.x >> 3, kvh = blockIdx.x & 7, d = threadIdx.x;
  __shared__ float red[128];
  __shared__ float sv[128];
  float v = kb[(size_t)m * (KVH_ * DHEAD) + kvh * DHEAD + d];
  red[d] = v * v; __syncthreads();
  for (int s = 64; s > 0; s >>= 1) { if (d < s) red[d] += red[d + s]; __syncthreads(); }
  float inv = rsqrtf(red[0] / (float)DHEAD + 1e-6f);
  float kn = kw[d] * v * inv;
  sv[d] = kn; __syncthreads();
  float rot = (d < 64) ? -sv[d + 64] : sv[d - 64];
  const int pos = PAST_ + m;
  float kr = kn * cosE[(size_t)pos * DHEAD + d] + rot * sinE[(size_t)pos * DHEAD + d];
  knew[((size_t)kvh * DHEAD + d) * NEW_T + m] = kr;
  vnew[((size_t)kvh * NEW_T + m) * DHEAD + d] =
      vb[(size_t)m * (KVH_ * DHEAD) + kvh * DHEAD + d];
}

// ---------------- attention scores: S(h,16,2064) = q_rot @ K + mask ----------
__global__ void attn_scores(const float* __restrict__ q, const float* __restrict__ kpast,
                            const float* __restrict__ knew, const float* __restrict__ causal,
                            const float* __restrict__ amask, float* __restrict__ S) {
  const int lane = threadIdx.x & 31;
  const int wave = blockIdx.x * (blockDim.x >> 5) + (threadIdx.x >> 5);
  const int h = wave / (ENDLEN / 16);
  const int t = wave % (ENDLEN / 16);
  if (h >= NHEAD) return;
  const int kvh = h >> 1;
  const int n = t * 16 + (lane & 15);
  const int arow = lane & 15;
  const int koff = (lane >> 4) << 1;
  const int rbase = (lane >> 4) << 3;
  v8f c = {};
  const float* Aq = q + (size_t)arow * (NHEAD * DHEAD) + h * DHEAD + koff;
  if (n < PAST_) {   // whole 16-tile is past (2048 % 16 == 0) -> wave-uniform
    const float* Bk = kpast + ((size_t)kvh * DHEAD + koff) * SMAX_ + n;
#pragma unroll 8
    for (int k = 0; k < DHEAD; k += 4) {
      v2f a, b;
      a.x = Aq[k];
      a.y = Aq[k + 1];
      b.x = Bk[(size_t)k * SMAX_];
      b.y = Bk[(size_t)(k + 1) * SMAX_];
      c = wmma4(a, b, c);
    }
  } else {
    const float* Bk = knew + ((size_t)kvh * DHEAD + koff) * NEW_T + (n - PAST_);
#pragma unroll 8
    for (int k = 0; k < DHEAD; k += 4) {
      v2f a, b;
      a.x = Aq[k];
      a.y = Aq[k + 1];
      b.x = Bk[(size_t)k * NEW_T];
      b.y = Bk[(size_t)(k + 1) * NEW_T];
      c = wmma4(a, b, c);
    }
  }
#pragma unroll
  for (int r = 0; r < 8; ++r) {
    int m = r + rbase;
    float mk = causal[(size_t)m * SMAX_ + n] * amask[(size_t)m * ENDLEN + n];
    S[((size_t)h * NEW_T + m) * ENDLEN + n] = c[r] + mk;
  }
}

// ---------------- row softmax over 2064 ----------------
__global__ void softmax_rows(float* __restrict__ S) {
  float* x = S + (size_t)blockIdx.x * ENDLEN;
  __shared__ float red[256];
  float mx = -3.4e38f;
  for (int j = threadIdx.x; j < ENDLEN; j += 256) mx = fmaxf(mx, x[j]);
  red[threadIdx.x] = mx; __syncthreads();
  for (int s = 128; s > 0; s >>= 1) {
    if ((int)threadIdx.x < s) red[threadIdx.x] = fmaxf(red[threadIdx.x], red[threadIdx.x + s]);
    __syncthreads();
  }
  mx = red[0]; __syncthreads();
  float sum = 0.f;
  for (int j = threadIdx.x; j < ENDLEN; j += 256) { float e = expf(x[j] - mx); x[j] = e; sum += e; }
  red[threadIdx.x] = sum; __syncthreads();
  for (int s = 128; s > 0; s >>= 1) {
    if ((int)threadIdx.x < s) red[threadIdx.x] += red[threadIdx.x + s];
    __syncthreads();
  }
  float inv = 1.f / red[0];
  for (int j = threadIdx.x; j < ENDLEN; j += 256) x[j] *= inv;
}

// ---- attn output chunk KC: PO[KC](16, NH*D) partial of P @ V ----
// Chunk bounds are compile-time; the past loop unrolls by 8, the new loop fully.
// grid: 16 blocks of 8 waves
template <int KC>
__global__ void attn_out_chunk(const float* __restrict__ S, const float* __restrict__ vpast,
                               const float* __restrict__ vnew, float* __restrict__ PO) {
  const int lane = threadIdx.x & 31;
  const int wave = blockIdx.x * (blockDim.x >> 5) + (threadIdx.x >> 5);
  const int h = wave >> 3, dt = wave & 7;
  if (h >= NHEAD) return;
  const int kvh = h >> 1;
  const int n = dt * 16 + (lane & 15);
  const int arow = lane & 15;
  const int koff = (lane >> 4) << 1;
  const int rbase = (lane >> 4) << 3;
  constexpr int sbeg = KC * SCH, send = sbeg + SCH;
  constexpr int pend = (send < PAST_) ? send : PAST_;   // past part of this chunk
  constexpr int nbeg = (sbeg > PAST_) ? sbeg : PAST_;   // new  part of this chunk
  v8f c = {};
  const float* As = S + (size_t)h * NEW_T * ENDLEN + (size_t)arow * ENDLEN + koff;
  const float* Vp = vpast + (size_t)kvh * SMAX_ * DHEAD + (size_t)koff * DHEAD + n;
  const float* Vn = vnew + (size_t)kvh * NEW_T * DHEAD + (size_t)koff * DHEAD + n;
#pragma unroll 8
  for (int s = sbeg; s < pend; s += 4) {
    v2f a, b;
    a.x = As[s];
    a.y = As[s + 1];
    b.x = Vp[(size_t)s * DHEAD];
    b.y = Vp[(size_t)(s + 1) * DHEAD];
    c = wmma4(a, b, c);
  }
#pragma unroll
  for (int s = nbeg; s < send; s += 4) {
    v2f a, b;
    a.x = As[s];
    a.y = As[s + 1];
    b.x = Vn[(size_t)(s - PAST_) * DHEAD];
    b.y = Vn[(size_t)(s - PAST_ + 1) * DHEAD];
    c = wmma4(a, b, c);
  }
  float* Pc = PO + (size_t)KC * (NEW_T * NHEAD * DHEAD);
#pragma unroll
  for (int r = 0; r < 8; ++r)
    Pc[(size_t)(r + rbase) * (NHEAD * DHEAD) + h * DHEAD + n] = c[r];
}

// ---------------- silu(g)*u ----------------
__global__ void silu_mul(const float* __restrict__ g, const float* __restrict__ u,
                         float* __restrict__ act) {
  int idx = blockIdx.x * blockDim.x + threadIdx.x;
  float x = g[idx];
  act[idx] = (x / (1.f + expf(-x))) * u[idx];
}

// ---------------- LM head GEMV + per-block argmax ----------------
__global__ void lm_gemv_argmax(const float* __restrict__ fh, const float* __restrict__ Wlm,
                               float* __restrict__ bmv, int* __restrict__ bmi) {
  int col = blockIdx.x * 256 + threadIdx.x;
  float acc = -3.4e38f;
  if (col < VOCAB) {
    float s = 0.f;
#pragma unroll 4
    for (int k = 0; k < HDIM; ++k) s += fh[k] * Wlm[(size_t)k * VOCAB + col];
    acc = s;
  }
  __shared__ float sv[256];
  __shared__ int si[256];
  sv[threadIdx.x] = acc; si[threadIdx.x] = col; __syncthreads();
  for (int s = 128; s > 0; s >>= 1) {
    if ((int)threadIdx.x < s) {
      float v2 = sv[threadIdx.x + s]; int i2 = si[threadIdx.x + s];
      float v1 = sv[threadIdx.x];     int i1 = si[threadIdx.x];
      if (v2 > v1 || (v2 == v1 && i2 < i1)) { sv[threadIdx.x] = v2; si[threadIdx.x] = i2; }
    }
    __syncthreads();
  }
  if (threadIdx.x == 0) { bmv[blockIdx.x] = sv[0]; bmi[blockIdx.x] = si[0]; }
}

__global__ void argmax_final(const float* __restrict__ bmv, const int* __restrict__ bmi,
                             int nb, int* __restrict__ out) {
  __shared__ float sv[256];
  __shared__ int si[256];
  float v = -3.4e38f; int idx = 0x7fffffff;
  if ((int)threadIdx.x < nb) { v = bmv[threadIdx.x]; idx = bmi[threadIdx.x]; }
  sv[threadIdx.x] = v; si[threadIdx.x] = idx; __syncthreads();
  for (int s = 128; s > 0; s >>= 1) {
    if ((int)threadIdx.x < s) {
      float v2 = sv[threadIdx.x + s]; int i2 = si[threadIdx.x + s];
      float v1 = sv[threadIdx.x];     int i1 = si[threadIdx.x];
      if (v2 > v1 || (v2 == v1 && i2 < i1)) { sv[threadIdx.x] = v2; si[threadIdx.x] = i2; }
    }
    __syncthreads();
  }
  if (threadIdx.x == 0) out[0] = si[0];
}

extern "C" void kernel_launch(void* const* d_in, const int* in_sizes, int n_in,
                              void* d_out, int out_size, void* d_ws, size_t ws_size,
                              hipStream_t stream) {
  const float*   key_buffer   = (const float*)d_in[0];
  const float*   value_buffer = (const float*)d_in[1];
  const int*     input_ids    = (const int*)d_in[2];
  const float*   amask        = (const float*)d_in[3];
  /* d_in[4] past_len: fixed 2048 (module constant) */
  const uint8_t* embd         = (const uint8_t*)d_in[5];
  const float*   scale        = (const float*)d_in[6];
  const float*   zp           = (const float*)d_in[7];
  const float*   cosE         = (const float*)d_in[8];
  const float*   sinE         = (const float*)d_in[9];
  const float*   causal       = (const float*)d_in[10];
  const float*   ln1          = (const float*)d_in[11];
  const float*   qw           = (const float*)d_in[12];
  const float*   kw           = (const float*)d_in[13];
  const float*   Wq           = (const float*)d_in[14];
  const float*   Wk           = (const float*)d_in[15];
  const float*   Wv           = (const float*)d_in[16];
  const float*   Wo           = (const float*)d_in[17];
  const float*   ln2          = (const float*)d_in[18];
  const float*   Wg           = (const float*)d_in[19];
  const float*   Wu           = (const float*)d_in[20];
  const float*   Wd           = (const float*)d_in[21];
  const float*   fnw          = (const float*)d_in[22];
  const float*   Wlm          = (const float*)d_in[23];

  float* ws   = (float*)d_ws;
  float* h    = ws;                              // 16*1024
  float* hn   = h    + NEW_T * HDIM;             // 16*1024
  float* q    = hn   + NEW_T * HDIM;             // 16*2048
  float* kb   = q    + NEW_T * NHEAD * DHEAD;    // 16*1024
  float* vb   = kb   + NEW_T * KVH_ * DHEAD;     // 16*1024
  float* knew = vb   + NEW_T * KVH_ * DHEAD;     // 8*128*16
  float* vnew = knew + KVH_ * DHEAD * NEW_T;     // 8*16*128
  float* S    = vnew + KVH_ * NEW_T * DHEAD;     // 16*16*2064
  float* ao   = S    + NHEAD * NEW_T * ENDLEN;   // 16*2048
  float* g    = ao   + NEW_T * NHEAD * DHEAD;    // 16*3072
  float* u    = g    + NEW_T * FF;               // 16*3072
  float* act  = u    + NEW_T * FF;               // 16*3072
  float* fh   = act  + NEW_T * FF;               // 1024
  float* bmv  = fh   + HDIM;                     // 256
  int*   bmi  = (int*)(bmv + 256);               // 256 ints
  float* P    = bmv  + 512;                      // KCH * 16 * 3072 max (1.5 MB)

  embed_k<<<(NEW_T * HDIM) / 256, 256, 0, stream>>>(input_ids, embd, scale, zp, h);

  for (int i = 0; i < NLAYER; ++i) {
    rms_rows<<<NEW_T, 256, 0, stream>>>(h, ln1 + (size_t)i * HDIM, hn, HDIM);

    // Q = hn @ Wq  (K=1024, N=2048) -- 1024 waves via split-K
    gemm16_splitk<HDIM / KCH><<<dim3((NHEAD * DHEAD) / 128, KCH), 256, 0, stream>>>(
        hn, Wq + (size_t)i * HDIM * NHEAD * DHEAD, P, NHEAD * DHEAD, HDIM, NHEAD * DHEAD);
    reduce_partials<<<(NEW_T * NHEAD * DHEAD) / 256, 256, 0, stream>>>(P, q, NEW_T * NHEAD * DHEAD, KCH, 0);
    // K = hn @ Wk  (K=1024, N=1024)
    gemm16_splitk<HDIM / KCH><<<dim3((KVH_ * DHEAD) / 128, KCH), 256, 0, stream>>>(
        hn, Wk + (size_t)i * HDIM * KVH_ * DHEAD, P, KVH_ * DHEAD, HDIM, KVH_ * DHEAD);
    reduce_partials<<<(NEW_T * KVH_ * DHEAD) / 256, 256, 0, stream>>>(P, kb, NEW_T * KVH_ * DHEAD, KCH, 0);
    // V = hn @ Wv
    gemm16_splitk<HDIM / KCH><<<dim3((KVH_ * DHEAD) / 128, KCH), 256, 0, stream>>>(
        hn, Wv + (size_t)i * HDIM * KVH_ * DHEAD, P, KVH_ * DHEAD, HDIM, KVH_ * DHEAD);
    reduce_partials<<<(NEW_T * KVH_ * DHEAD) / 256, 256, 0, stream>>>(P, vb, NEW_T * KVH_ * DHEAD, KCH, 0);

    qnorm_rope<<<NEW_T * NHEAD, DHEAD, 0, stream>>>(q, qw + (size_t)i * DHEAD, cosE, sinE);
    kv_rope<<<NEW_T * KVH_, DHEAD, 0, stream>>>(kb, vb, kw + (size_t)i * DHEAD,
                                                cosE, sinE, knew, vnew);

    attn_scores<<<(NHEAD * (ENDLEN / 16)) / 8, 256, 0, stream>>>(
        q, key_buffer + (size_t)i * KVH_ * DHEAD * SMAX_, knew, causal, amask, S);
    softmax_rows<<<NHEAD * NEW_T, 256, 0, stream>>>(S);

    const float* Vl = value_buffer + (size_t)i * KVH_ * SMAX_ * DHEAD;
    attn_out_chunk<0><<<(NHEAD * 8) / 8, 256, 0, stream>>>(S, Vl, vnew, P);
    attn_out_chunk<1><<<(NHEAD * 8) / 8, 256, 0, stream>>>(S, Vl, vnew, P);
    attn_out_chunk<2><<<(NHEAD * 8) / 8, 256, 0, stream>>>(S, Vl, vnew, P);
    attn_out_chunk<3><<<(NHEAD * 8) / 8, 256, 0, stream>>>(S, Vl, vnew, P);
    reduce_partials<<<(NEW_T * NHEAD * DHEAD) / 256, 256, 0, stream>>>(P, ao, NEW_T * NHEAD * DHEAD, 4, 0);

    // h += ao @ Wo  (K=2048, N=1024) -- 512 waves via split-K
    gemm16_splitk<(NHEAD * DHEAD) / KCH><<<dim3(HDIM / 128, KCH), 256, 0, stream>>>(
        ao, Wo + (size_t)i * NHEAD * DHEAD * HDIM, P, HDIM, NHEAD * DHEAD, HDIM);
    reduce_partials<<<(NEW_T * HDIM) / 256, 256, 0, stream>>>(P, h, NEW_T * HDIM, KCH, 1);

    rms_rows<<<NEW_T, 256, 0, stream>>>(h, ln2 + (size_t)i * HDIM, hn, HDIM);
    // gate / up (K=1024, N=3072) -- 1536 waves each
    gemm16_splitk<HDIM / KCH><<<dim3(FF / 128, KCH), 256, 0, stream>>>(
        hn, Wg + (size_t)i * HDIM * FF, P, FF, HDIM, FF);
    reduce_partials<<<(NEW_T * FF) / 256, 256, 0, stream>>>(P, g, NEW_T * FF, KCH, 0);
    gemm16_splitk<HDIM / KCH><<<dim3(FF / 128, KCH), 256, 0, stream>>>(
        hn, Wu + (size_t)i * HDIM * FF, P, FF, HDIM, FF);
    reduce_partials<<<(NEW_T * FF) / 256, 256, 0, stream>>>(P, u, NEW_T * FF, KCH, 0);

    silu_mul<<<(NEW_T * FF) / 256, 256, 0, stream>>>(g, u, act);

    // h += act @ Wdown (K=3072, N=1024) -- 512 waves via split-K
    gemm16_splitk<FF / KCH><<<dim3(HDIM / 128, KCH), 256, 0, stream>>>(
        act, Wd + (size_t)i * FF * HDIM, P, HDIM, FF, HDIM);
    reduce_partials<<<(NEW_T * HDIM) / 256, 256, 0, stream>>>(P, h, NEW_T * HDIM, KCH, 1);
  }

  rms_rows<<<1, 256, 0, stream>>>(h + (size_t)(NEW_T - 1) * HDIM, fnw, fh, HDIM);
  int nb = (VOCAB + 255) / 256;  // 197
  lm_gemv_argmax<<<nb, 256, 0, stream>>>(fh, Wlm, bmv, bmi);
  argmax_final<<<1, 256, 0, stream>>>(bmv, bmi, nb, (int*)d_out);
}